// trunc_conv_abs_73985106641587
// MI455X (gfx1250) — compile-verified
//
#include <hip/hip_runtime.h>
#include <hip/hip_bf16.h>
#include <math.h>

typedef __attribute__((ext_vector_type(2))) float v2f;
typedef __attribute__((ext_vector_type(8))) float v8f;

#define L 512
#define LL (L*L)           // 262144 pixels per image
#define OD 508
#define OOD (OD*OD)        // 258064 outputs per image
#define KTOP 100u
#define NIMG 256

// ---------------------------------------------------------------------------
// zero the per-image MSB histograms
// ---------------------------------------------------------------------------
__global__ void zero_hist_kernel(unsigned* __restrict__ g) {
    int i = blockIdx.x * blockDim.x + threadIdx.x;
    if (i < NIMG * 256) g[i] = 0u;
}

// ---------------------------------------------------------------------------
// pass1: per 32x32 tile compute r = (x*S - T)*count, emit sortable key
//        |r| as float bits, fused per-image 256-bin MSB histogram.
//        T (5x5 clipped box of kern_avg) column pass done via WMMA banded GEMM.
// ---------------------------------------------------------------------------
__global__ __launch_bounds__(128)
void pass1_rkey(const float* __restrict__ x, const float* __restrict__ w,
                unsigned* __restrict__ rkey, unsigned* __restrict__ ghist) {
  __shared__ float xs[40][41];    // x tile with 4-halo
  __shared__ float Ks[36][37];    // kern_avg tile (zero outside [0,507]^2)
  __shared__ float Rs[36][33];    // row box sums
  __shared__ float wsm[5][5];
  __shared__ float P[6][6];       // 2D inclusive prefix sums of w (for S)
  __shared__ unsigned hist[256];

  const int t   = threadIdx.x;
  const int img = blockIdx.z;
  const int ty0 = blockIdx.y * 32;
  const int tx0 = blockIdx.x * 32;
  const int gy0 = ty0 - 4, gx0 = tx0 - 4;
  const float* xi = x + (size_t)img * LL;

  if (t < 25) wsm[t / 5][t % 5] = w[t];
  for (int i = t; i < 256; i += 128) hist[i] = 0u;
  __syncthreads();
  if (t == 0) {
    for (int j = 0; j < 6; ++j) P[0][j] = 0.f;
    for (int i = 1; i < 6; ++i) {
      P[i][0] = 0.f;
      float row = 0.f;
      for (int j = 1; j < 6; ++j) { row += wsm[i-1][j-1]; P[i][j] = P[i-1][j] + row; }
    }
  }
  // load 40x40 x tile (zero outside image)
  for (int i = t; i < 1600; i += 128) {
    int rr = i / 40, cc = i % 40;
    int gy = gy0 + rr, gx = gx0 + cc;
    float v = 0.f;
    if ((unsigned)gy < 512u && (unsigned)gx < 512u) v = xi[gy * 512 + gx];
    xs[rr][cc] = v;
  }
  __syncthreads();
  // kern_avg tile: 36x36, zero where outside valid conv domain [0,507]^2
  for (int i = t; i < 1296; i += 128) {
    int a = i / 36, b = i % 36;
    int ay = gy0 + a, bx2 = gx0 + b;
    float acc = 0.f;
    #pragma unroll
    for (int ki = 0; ki < 5; ++ki)
      #pragma unroll
      for (int kj = 0; kj < 5; ++kj)
        acc += xs[a + ki][b + kj] * wsm[ki][kj];
    bool ok = ((unsigned)ay < 508u) && ((unsigned)bx2 < 508u);
    Ks[a][b] = ok ? acc * 0.04f : 0.f;   // /25 folded in; zeroing handles clipping
  }
  __syncthreads();
  // horizontal 5-tap box: Rs[a][nl] = sum_{d=0..4} Ks[a][nl+d]
  for (int i = t; i < 1152; i += 128) {
    int a = i / 32, nl = i % 32;
    Rs[a][nl] = Ks[a][nl] + Ks[a][nl+1] + Ks[a][nl+2] + Ks[a][nl+3] + Ks[a][nl+4];
  }
  __syncthreads();
  // vertical 5-tap box as banded GEMM: T_sub(16x16) = Band(16x20) * Rs(20x16)
  // 5 chained V_WMMA_F32_16X16X4_F32 per wave, one 16x16 subtile per wave.
  const int wave = t >> 5, lane = t & 31, half = lane >> 4, ln = lane & 15;
  const int ms = (wave >> 1) * 16, ns = (wave & 1) * 16;
  v8f acc;
  #pragma unroll
  for (int i = 0; i < 8; ++i) acc[i] = 0.f;
  const int k0 = 2 * half;                 // f32 A/B frag: lanes<16 hold K=0,1; lanes>=16 hold K=2,3
  #pragma unroll
  for (int c = 0; c < 5; ++c) {
    int a0 = 4 * c + k0, a1 = a0 + 1;
    v2f Af, Bf;
    Af[0] = (a0 >= ln && a0 <= ln + 4) ? 1.f : 0.f;   // Band[m=ln][a0]
    Af[1] = (a1 >= ln && a1 <= ln + 4) ? 1.f : 0.f;
    Bf[0] = Rs[ms + a0][ns + ln];                      // B[k][n=ln]
    Bf[1] = Rs[ms + a1][ns + ln];
    acc = __builtin_amdgcn_wmma_f32_16x16x4_f32(false, Af, false, Bf,
                                                (short)0, acc, false, false);
  }
  // D layout: VGPR d -> row d (lanes 0-15) / row d+8 (lanes 16-31), col = ln
  #pragma unroll
  for (int d = 0; d < 8; ++d) {
    int mp = d + 8 * half;
    int y  = ty0 + ms + mp;
    int xg = tx0 + ns + ln;
    float T  = acc[d];
    float xv = xs[4 + ms + mp][4 + ns + ln];
    int ilo = max(0, y - 507),  ihi = min(4, y);
    int jlo = max(0, xg - 507), jhi = min(4, xg);
    float S = P[ihi+1][jhi+1] - P[ilo][jhi+1] - P[ihi+1][jlo] + P[ilo][jlo];
    int cy = min(y, 507)  - max(y  - 4, 0) + 1;
    int cx = min(xg, 507) - max(xg - 4, 0) + 1;
    float r = (xv * S - T) * (float)(cy * cx);
    unsigned key = __float_as_uint(fabsf(r));   // monotone since |r| >= 0
    rkey[(size_t)img * LL + y * 512 + xg] = key;
    atomicAdd(&hist[key >> 24], 1u);
  }
  __syncthreads();
  for (int i = t; i < 256; i += 128)
    if (hist[i]) atomicAdd(&ghist[img * 256 + i], hist[i]);
}

// ---------------------------------------------------------------------------
// pick the MSB digit of the 100th-largest key from the fused histogram
// ---------------------------------------------------------------------------
__global__ __launch_bounds__(256)
void select8_kernel(const unsigned* __restrict__ ghist, uint2* __restrict__ state) {
  int img = threadIdx.x;
  unsigned k = KTOP, pref = 0u;
  for (int b = 255; b >= 0; --b) {
    unsigned c = ghist[img * 256 + b];
    if (c >= k) { pref = ((unsigned)b) << 24; break; }
    k -= c;
  }
  state[img] = make_uint2(pref, k);
}

// ---------------------------------------------------------------------------
// one MSD radix-select refinement pass (8-bit digit at `shift`)
// ---------------------------------------------------------------------------
__global__ __launch_bounds__(256)
void radix_pass(const unsigned* __restrict__ rkey, uint2* __restrict__ state, int shift) {
  __shared__ unsigned h[256];
  const int img = blockIdx.x, t = threadIdx.x;
  h[t] = 0u;
  uint2 st = state[img];
  const unsigned prefHigh = st.x >> (shift + 8);
  __syncthreads();
  const uint4* k4 = (const uint4*)(rkey + (size_t)img * LL);
  for (int i = t; i < LL / 4; i += 256) {
    uint4 kk = k4[i];
    if ((kk.x >> (shift + 8)) == prefHigh) atomicAdd(&h[(kk.x >> shift) & 255], 1u);
    if ((kk.y >> (shift + 8)) == prefHigh) atomicAdd(&h[(kk.y >> shift) & 255], 1u);
    if ((kk.z >> (shift + 8)) == prefHigh) atomicAdd(&h[(kk.z >> shift) & 255], 1u);
    if ((kk.w >> (shift + 8)) == prefHigh) atomicAdd(&h[(kk.w >> shift) & 255], 1u);
  }
  __syncthreads();
  if (t == 0) {
    unsigned k = st.y;
    for (int b = 255; b >= 0; --b) {
      unsigned c = h[b];
      if (c >= k) { st.x |= ((unsigned)b) << shift; st.y = k; break; }
      k -= c;
    }
    state[img] = st;   // after shift==0 pass, st.x == exact 100th-largest key
  }
}

// ---------------------------------------------------------------------------
// final masked 5x5 conv: z = (key < thr), out = conv(x*z, w) + bias
// ---------------------------------------------------------------------------
__global__ __launch_bounds__(256)
void pass5_conv(const float* __restrict__ x, const float* __restrict__ w,
                const float* __restrict__ bias, const unsigned* __restrict__ rkey,
                const uint2* __restrict__ state, float* __restrict__ out) {
  __shared__ float xm[36][37];
  __shared__ float wsm[25];
  const int t = threadIdx.x, img = blockIdx.z;
  const int ty0 = blockIdx.y * 32, tx0 = blockIdx.x * 32;
  const unsigned thr = state[img].x;       // uniform scalar load
  const float bv = bias[0];
  if (t < 25) wsm[t] = w[t];
  const float*    xi = x    + (size_t)img * LL;
  const unsigned* ki = rkey + (size_t)img * LL;
  for (int i = t; i < 1296; i += 256) {
    int rr = i / 36, cc = i % 36;
    int gy = ty0 + rr, gx = tx0 + cc;
    float v = 0.f;
    if (gy < 512 && gx < 512) {
      int idx = gy * 512 + gx;
      v = (ki[idx] < thr) ? xi[idx] : 0.f;   // top-100 positions zeroed
    }
    xm[rr][cc] = v;
  }
  __syncthreads();
  const int nl = t & 31, mb = (t >> 5) * 4;
  #pragma unroll
  for (int dm = 0; dm < 4; ++dm) {
    int m = mb + dm, y = ty0 + m, xg = tx0 + nl;
    if (y < OD && xg < OD) {
      float acc = bv;
      #pragma unroll
      for (int i2 = 0; i2 < 5; ++i2)
        #pragma unroll
        for (int j2 = 0; j2 < 5; ++j2)
          acc += wsm[i2 * 5 + j2] * xm[m + i2][nl + j2];
      out[(size_t)img * OOD + y * OD + xg] = acc;
    }
  }
}

// ---------------------------------------------------------------------------
extern "C" void kernel_launch(void* const* d_in, const int* in_sizes, int n_in,
                              void* d_out, int out_size, void* d_ws, size_t ws_size,
                              hipStream_t stream) {
  const float* x    = (const float*)d_in[0];   // (256,1,512,512) f32
  const float* w    = (const float*)d_in[1];   // (1,1,5,5) f32
  const float* bias = (const float*)d_in[2];   // (1,) f32
  float* out = (float*)d_out;                  // (256,1,508,508) f32

  unsigned* rkey  = (unsigned*)d_ws;                    // 256*262144*4 = 268 MB
  unsigned* ghist = rkey + (size_t)NIMG * LL;           // 65536 * 4
  uint2*    state = (uint2*)(ghist + NIMG * 256);       // 256 * 8 (8B aligned)

  zero_hist_kernel<<<(NIMG * 256 + 255) / 256, 256, 0, stream>>>(ghist);
  pass1_rkey<<<dim3(16, 16, NIMG), 128, 0, stream>>>(x, w, rkey, ghist);
  select8_kernel<<<1, 256, 0, stream>>>(ghist, state);
  radix_pass<<<NIMG, 256, 0, stream>>>(rkey, state, 16);
  radix_pass<<<NIMG, 256, 0, stream>>>(rkey, state, 8);
  radix_pass<<<NIMG, 256, 0, stream>>>(rkey, state, 0);
  pass5_conv<<<dim3(16, 16, NIMG), 256, 0, stream>>>(x, w, bias, rkey, state, out);
}